// RenderNet_69355131895961
// MI455X (gfx1250) — compile-verified
//
#include <hip/hip_runtime.h>

#define ZF_THREADS 256
#define ZF_VEC 8  // float4 stores per thread

// ---------------------------------------------------------------------------
// Kernel 1: stream 192 MiB of zeros at HBM store bandwidth using the CDNA5
// async LDS->global store path (ASYNCcnt-tracked DMA, 16B/lane per instr).
// Zeros are staged once in LDS; every lane points at the same 16B of LDS
// (broadcast read) and streams them to its own global address.
// ---------------------------------------------------------------------------
__global__ void __launch_bounds__(ZF_THREADS)
zero_fill_async(float* __restrict__ out, unsigned long long total4) {
  __shared__ __align__(16) float lz[64];
  if (threadIdx.x < 64) lz[threadIdx.x] = 0.0f;
  __syncthreads();  // drain DScnt: LDS zeros visible before async DMA reads

  // ptrtoint of an LDS pointer yields the wave-relative LDS byte offset.
  const unsigned lds_addr = (unsigned)(unsigned long long)(&lz[0]);

  const unsigned long long blkBase =
      (unsigned long long)blockIdx.x * (ZF_THREADS * ZF_VEC);
  const unsigned long long base = blkBase + threadIdx.x;

  if (blkBase + (ZF_THREADS * ZF_VEC) <= total4) {
    // Fast path: whole block in range -> 8 unguarded async B128 stores.
    unsigned long long gaddr = (unsigned long long)(out + base * 4ull);
#pragma unroll
    for (int it = 0; it < ZF_VEC; ++it) {
      asm volatile("global_store_async_from_lds_b128 %0, %1, off"
                   :
                   : "v"(gaddr), "v"(lds_addr)
                   : "memory");
      gaddr += (unsigned long long)ZF_THREADS * 16ull;
    }
  } else {
    // Tail block: per-element guard.
#pragma unroll
    for (int it = 0; it < ZF_VEC; ++it) {
      unsigned long long q = base + (unsigned long long)it * ZF_THREADS;
      if (q < total4) {
        unsigned long long gaddr = (unsigned long long)(out + q * 4ull);
        asm volatile("global_store_async_from_lds_b128 %0, %1, off"
                     :
                     : "v"(gaddr), "v"(lds_addr)
                     : "memory");
      }
    }
  }
  // Drain ASYNCcnt before the wave retires so LDS is provably live for all
  // outstanding DMA reads (S_ENDPGM also wait-idles, but be explicit).
  asm volatile("s_wait_asynccnt 0x0" ::: "memory");
}

// ---------------------------------------------------------------------------
// Kernel 2: stamp the two 7x7 nested-square blobs (x 3 channels = 294 cells).
// Bit-faithful to the reference: v = z * (1.0f/z) in f32; kernel value
// K(m) = (m+1)/(center+1); template = clip(n_steps*blob0 + blob1, 0, 1).
// The two stamps are spatially disjoint, so each cell has one contribution.
// ---------------------------------------------------------------------------
__global__ void stamp_patches(const float* x0, const float* y0,
                              const float* x1, const float* y1,
                              const int* p_x0, const int* p_y0,
                              const int* p_x1, const int* p_y1,
                              const int* p_nsteps, const int* p_steep,
                              const int* p_imsize, const int* p_lw,
                              float* __restrict__ out) {
  const int lw = *p_lw;          // 7
  const int center = lw >> 1;    // 3
  const int imsize = *p_imsize;  // 4096
  const int steep = *p_steep;    // 0
  const float nstepsf = (float)(*p_nsteps);  // 3800.0f

  const int r0 = steep ? *p_y0 : *p_x0;  // 100
  const int c0 = steep ? *p_x0 : *p_y0;  // 200
  const int r1 = *p_x1;                  // 3900
  const int c1 = *p_y1;                  // 3500

  const float z0 = *x0 + *y0;
  const float v0 = z0 * (1.0f / z0);
  const float z1 = *x1 + *y1;
  const float v1 = z1 * (1.0f / z1);

  const int per = lw * lw;           // 49
  const int total = 2 * 3 * per;     // 294
  const int t = blockIdx.x * blockDim.x + threadIdx.x;
  if (t >= total) return;

  const int which = t / (3 * per);
  const int rem = t - which * 3 * per;
  const int chan = rem / per;
  const int cell = rem - chan * per;
  const int di = cell / lw;
  const int dj = cell - di * lw;

  // nested-square kernel value
  const int m = min(min(di, lw - 1 - di), min(dj, lw - 1 - dj));
  const float kval = (float)(m + 1) / (float)(center + 1);

  int row, col;
  float val;
  if (which == 0) {
    row = r0 + di - center;
    col = c0 + dj - center;
    val = nstepsf * (v0 * kval);  // n_steps * blobs[0]
  } else {
    row = r1 + di - center;
    col = c1 + dj - center;
    val = v1 * kval;              // + blobs[1]
  }
  val = fminf(fmaxf(val, 0.0f), 1.0f);  // jnp.clip(..., 0, 1)

  if (row >= 0 && row < imsize && col >= 0 && col < imsize) {
    const size_t plane = (size_t)imsize * (size_t)imsize;
    out[(size_t)chan * plane + (size_t)row * (size_t)imsize + (size_t)col] = val;
  }
}

// ---------------------------------------------------------------------------
// Inputs (dict order): x0,y0,x1,y1 (f32[1]); _x0,_y0,_x1,_y1,n_steps,steep,
// imsize,linewidth (i32[1]). Output: (1,3,imsize,imsize) f32, flat.
// ---------------------------------------------------------------------------
extern "C" void kernel_launch(void* const* d_in, const int* in_sizes, int n_in,
                              void* d_out, int out_size, void* d_ws,
                              size_t ws_size, hipStream_t stream) {
  (void)in_sizes; (void)n_in; (void)d_ws; (void)ws_size;
  float* out = (float*)d_out;

  const unsigned long long total4 = (unsigned long long)out_size / 4ull;
  const unsigned long long perBlock = (unsigned long long)ZF_THREADS * ZF_VEC;
  const unsigned int blocks = (unsigned int)((total4 + perBlock - 1) / perBlock);

  zero_fill_async<<<blocks, ZF_THREADS, 0, stream>>>(out, total4);

  stamp_patches<<<1, 512, 0, stream>>>(
      (const float*)d_in[0], (const float*)d_in[1], (const float*)d_in[2],
      (const float*)d_in[3], (const int*)d_in[4], (const int*)d_in[5],
      (const int*)d_in[6], (const int*)d_in[7], (const int*)d_in[8],
      (const int*)d_in[9], (const int*)d_in[10], (const int*)d_in[11], out);
}